// TorchscriptFeatureExtractor_2241972928788
// MI455X (gfx1250) — compile-verified
//
#include <hip/hip_runtime.h>
#include <stdint.h>
#include <math.h>

#define EPSF 1e-12f

typedef int v2i __attribute__((ext_vector_type(2)));
typedef int v4i __attribute__((ext_vector_type(4)));

struct F3 { float x, y, z; };

__device__ __forceinline__ F3 mkf3(float x, float y, float z) { F3 r; r.x = x; r.y = y; r.z = z; return r; }
__device__ __forceinline__ F3 f3sub(F3 a, F3 b) { return mkf3(a.x - b.x, a.y - b.y, a.z - b.z); }
__device__ __forceinline__ float f3dot(F3 a, F3 b) { return fmaf(a.x, b.x, fmaf(a.y, b.y, a.z * b.z)); }
__device__ __forceinline__ F3 f3cross(F3 a, F3 b) {
    return mkf3(fmaf(a.y, b.z, -(a.z * b.y)),
                fmaf(a.z, b.x, -(a.x * b.z)),
                fmaf(a.x, b.y, -(a.y * b.x)));
}

// Position gathers use default (RT) temporal hint on purpose: the 1.5 MB
// table should stay resident in WGP$/L2; all streaming accesses are NT.
__device__ __forceinline__ F3 ldpos(const float* __restrict__ p, int i) {
    const float* q = p + 3 * (size_t)i;
    return mkf3(q[0], q[1], q[2]);
}

// minimum image: frac = v @ IB ; frac -= round(frac) ; disp = frac @ B
// row-vector convention: out_j = sum_k v_k * M[k*3+j]
__device__ __forceinline__ F3 min_image(F3 v, const float* B, const float* IB) {
    float fx = fmaf(v.x, IB[0], fmaf(v.y, IB[3], v.z * IB[6]));
    float fy = fmaf(v.x, IB[1], fmaf(v.y, IB[4], v.z * IB[7]));
    float fz = fmaf(v.x, IB[2], fmaf(v.y, IB[5], v.z * IB[8]));
    fx -= rintf(fx); fy -= rintf(fy); fz -= rintf(fz);   // round-half-even == jnp.round
    return mkf3(fmaf(fx, B[0], fmaf(fy, B[3], fz * B[6])),
                fmaf(fx, B[1], fmaf(fy, B[4], fz * B[7])),
                fmaf(fx, B[2], fmaf(fy, B[5], fz * B[8])));
}

// CDNA5 async path: stage one i32 (global -> LDS slot for this lane), tracked
// by ASYNCcnt; drained with s_wait_asynccnt after the compute chain. NT hint:
// the scatter-index stream is touch-once, keep it out of the near cache.
__device__ __forceinline__ void async_stage_i32(const int* gsrc, int* lds_slot) {
    unsigned long long ga = (unsigned long long)(uintptr_t)gsrc;
    unsigned int la = (unsigned int)(uintptr_t)lds_slot;
    asm volatile("global_load_async_to_lds_b32 %0, %1, off th:TH_LOAD_NT"
                 :: "v"(la), "v"(ga) : "memory");
}

__device__ __forceinline__ void load_box(const float* __restrict__ bi, float* B, float* IB) {
    #pragma unroll
    for (int k = 0; k < 9; ++k) { B[k] = bi[k]; IB[k] = bi[9 + k]; }
}

// fused NT weight-load + NT scatter-store of the final feature value
__device__ __forceinline__ void emit(float val, int p,
                                     const float* __restrict__ wts,
                                     float* __restrict__ out) {
    float w = __builtin_nontemporal_load(wts + p);
    __builtin_nontemporal_store(val * w, out + p);
}

// one-thread prep: ws[0..8] = box, ws[9..17] = inv(box)
__global__ void invbox_kernel(const float* __restrict__ box, float* __restrict__ ws) {
    if (threadIdx.x == 0 && blockIdx.x == 0) {
        float a = box[0], b = box[1], c = box[2];
        float d = box[3], e = box[4], f = box[5];
        float g = box[6], h = box[7], i = box[8];
        float A0 = e * i - f * h, A1 = c * h - b * i, A2 = b * f - c * e;
        float A3 = f * g - d * i, A4 = a * i - c * g, A5 = c * d - a * f;
        float A6 = d * h - e * g, A7 = b * g - a * h, A8 = a * e - b * d;
        float rdet = 1.0f / (a * A0 + b * A3 + c * A6);
        #pragma unroll
        for (int k = 0; k < 9; ++k) ws[k] = box[k];
        ws[9]  = A0 * rdet; ws[10] = A1 * rdet; ws[11] = A2 * rdet;
        ws[12] = A3 * rdet; ws[13] = A4 * rdet; ws[14] = A5 * rdet;
        ws[15] = A6 * rdet; ws[16] = A7 * rdet; ws[17] = A8 * rdet;
    }
}

// ---------------- distances ----------------
__global__ __launch_bounds__(256) void distance_kernel(
    const float* __restrict__ pos, const float* __restrict__ bi,
    const float* __restrict__ wts, const int* __restrict__ pair,
    const int* __restrict__ fpos, float* __restrict__ out, int n)
{
    __shared__ int s_fp[256];
    const int i = blockIdx.x * 256 + threadIdx.x;
    if (i >= n) return;

    async_stage_i32(fpos + i, &s_fp[threadIdx.x]);

    float B[9], IB[9];
    load_box(bi, B, IB);

    v2i pr = __builtin_nontemporal_load(((const v2i*)pair) + i);   // b64 NT, coalesced
    F3 dv = min_image(f3sub(ldpos(pos, pr.y), ldpos(pos, pr.x)), B, IB);
    float val = sqrtf(fmaxf(f3dot(dv, dv), EPSF));

    __builtin_amdgcn_s_wait_asynccnt(0);
    emit(val, s_fp[threadIdx.x], wts, out);               // ds_load_b32 + NT store
}

// ---------------- angles ----------------
__global__ __launch_bounds__(256) void angle_kernel(
    const float* __restrict__ pos, const float* __restrict__ bi,
    const float* __restrict__ wts, const int* __restrict__ trip,
    const int* __restrict__ fpos, float* __restrict__ out, int n)
{
    __shared__ int s_fp[256];
    const int i = blockIdx.x * 256 + threadIdx.x;
    if (i >= n) return;

    async_stage_i32(fpos + i, &s_fp[threadIdx.x]);

    float B[9], IB[9];
    load_box(bi, B, IB);

    const int* t = trip + 3 * (size_t)i;                  // 12B stream, NT
    int t0 = __builtin_nontemporal_load(t + 0);
    int t1 = __builtin_nontemporal_load(t + 1);
    int t2 = __builtin_nontemporal_load(t + 2);
    F3 pc = ldpos(pos, t1);
    F3 v1 = min_image(f3sub(ldpos(pos, t0), pc), B, IB);
    F3 v2 = min_image(f3sub(ldpos(pos, t2), pc), B, IB);
    float n1 = sqrtf(fmaxf(f3dot(v1, v1), EPSF));
    float n2 = sqrtf(fmaxf(f3dot(v2, v2), EPSF));
    float ct = f3dot(v1, v2) / (n1 * n2);
    ct = fminf(1.0f, fmaxf(-1.0f, ct));
    float val = acosf(ct);

    __builtin_amdgcn_s_wait_asynccnt(0);
    emit(val, s_fp[threadIdx.x], wts, out);
}

// ---------------- dihedrals ----------------
__global__ __launch_bounds__(256) void dihedral_kernel(
    const float* __restrict__ pos, const float* __restrict__ bi,
    const float* __restrict__ wts, const int* __restrict__ quad,
    const int* __restrict__ fpos, float* __restrict__ out, int n)
{
    __shared__ int s_fp[256];
    const int i = blockIdx.x * 256 + threadIdx.x;
    if (i >= n) return;

    async_stage_i32(fpos + i, &s_fp[threadIdx.x]);

    float B[9], IB[9];
    load_box(bi, B, IB);

    v4i q = __builtin_nontemporal_load(((const v4i*)quad) + i);    // b128 NT, coalesced
    F3 p0 = ldpos(pos, q.x), p1 = ldpos(pos, q.y);
    F3 p2 = ldpos(pos, q.z), p3 = ldpos(pos, q.w);
    F3 b0 = min_image(f3sub(p1, p0), B, IB);
    F3 b1 = min_image(f3sub(p2, p1), B, IB);
    F3 b2 = min_image(f3sub(p3, p2), B, IB);
    F3 c0 = f3cross(b0, b1);
    F3 c1 = f3cross(b1, b2);
    float s0 = rsqrtf(fmaxf(f3dot(c0, c0), EPSF));
    float s1 = rsqrtf(fmaxf(f3dot(c1, c1), EPSF));
    float sb = rsqrtf(fmaxf(f3dot(b1, b1), EPSF));
    c0 = mkf3(c0.x * s0, c0.y * s0, c0.z * s0);
    c1 = mkf3(c1.x * s1, c1.y * s1, c1.z * s1);
    F3 b1u = mkf3(b1.x * sb, b1.y * sb, b1.z * sb);
    float x = f3dot(c0, c1);
    float y = f3dot(f3cross(c0, c1), b1u);
    bool ok = (fabsf(x) + fabsf(y)) >= EPSF;
    float ang = atan2f(ok ? y : 0.0f, ok ? x : 1.0f);
    float val = ok ? ang : 0.0f;

    __builtin_amdgcn_s_wait_asynccnt(0);
    emit(val, s_fp[threadIdx.x], wts, out);
}

extern "C" void kernel_launch(void* const* d_in, const int* in_sizes, int n_in,
                              void* d_out, int out_size, void* d_ws, size_t ws_size,
                              hipStream_t stream) {
    const float* pos   = (const float*)d_in[0];
    const float* box   = (const float*)d_in[1];
    const float* wts   = (const float*)d_in[2];
    const int*   dpair = (const int*)d_in[3];
    const int*   dposi = (const int*)d_in[4];
    const int*   atrip = (const int*)d_in[5];
    const int*   aposi = (const int*)d_in[6];
    const int*   quad  = (const int*)d_in[7];
    const int*   qposi = (const int*)d_in[8];

    const int D = in_sizes[4];   // distance_positions: D elements
    const int A = in_sizes[6];   // angle_positions:    A elements
    const int Q = in_sizes[8];   // dihedral_positions: Q elements

    float* ws  = (float*)d_ws;   // 18 floats: box + inv_box
    float* out = (float*)d_out;

    invbox_kernel<<<1, 1, 0, stream>>>(box, ws);

    distance_kernel<<<(D + 255) / 256, 256, 0, stream>>>(pos, ws, wts, dpair, dposi, out, D);
    angle_kernel   <<<(A + 255) / 256, 256, 0, stream>>>(pos, ws, wts, atrip, aposi, out, A);
    dihedral_kernel<<<(Q + 255) / 256, 256, 0, stream>>>(pos, ws, wts, quad,  qposi, out, Q);
}